// EventDiffusion_27779848470659
// MI455X (gfx1250) — compile-verified
//
#include <hip/hip_runtime.h>
#include <math.h>

typedef __attribute__((ext_vector_type(2))) float v2f;
typedef __attribute__((ext_vector_type(8))) float v8f;

#define D 256

// ---------------------------------------------------------------------------
// GEMM: Y[M x 256] = X[M x 256] @ W[256 x 256]  (fp32 WMMA 16x16x4)
// Block = 128 threads (4 waves). Block covers a 16(M) x 256(N) slab.
// Each wave owns 4 adjacent 16x16 N-tiles (4 independent accumulators).
// A layout (16x4 f32): lane l(0-15) row M=l; VGPR0=K(k0+2*hi), VGPR1=K(k0+1+2*hi)
// B layout (4x16 f32): lanes 0-15 col N; VGPR0 row K=k0+2*hi, VGPR1 row k0+1+2*hi
// C layout (16x16 f32): VGPR r -> row r + 8*hi, col = lane&15
// ---------------------------------------------------------------------------
__global__ void __launch_bounds__(128)
gemm256_wmma_f32(const float* __restrict__ X, const float* __restrict__ W,
                 float* __restrict__ Y) {
  const int wave  = threadIdx.x >> 5;
  const int lane  = threadIdx.x & 31;
  const int l15   = lane & 15;
  const int hi    = lane >> 4;           // 0 or 1 (half-wave)
  const int mBase = blockIdx.x * 16;
  const int nBase = wave * 64;

  v8f acc0 = {}, acc1 = {}, acc2 = {}, acc3 = {};
  const float* xr = X + (size_t)(mBase + l15) * D;
  const int c0 = nBase + l15;

#pragma unroll 4
  for (int k0 = 0; k0 < D; k0 += 4) {
    const int ka = k0 + (hi << 1);
    v2f a = *(const v2f*)(xr + ka);
    const float* w0 = W + (size_t)ka * D;     // row ka ; row ka+1 at +D
    v2f b0 = { w0[c0 +  0], w0[D + c0 +  0] };
    v2f b1 = { w0[c0 + 16], w0[D + c0 + 16] };
    v2f b2 = { w0[c0 + 32], w0[D + c0 + 32] };
    v2f b3 = { w0[c0 + 48], w0[D + c0 + 48] };
    acc0 = __builtin_amdgcn_wmma_f32_16x16x4_f32(false, a, false, b0, (short)0, acc0, false, false);
    acc1 = __builtin_amdgcn_wmma_f32_16x16x4_f32(false, a, false, b1, (short)0, acc1, false, false);
    acc2 = __builtin_amdgcn_wmma_f32_16x16x4_f32(false, a, false, b2, (short)0, acc2, false, false);
    acc3 = __builtin_amdgcn_wmma_f32_16x16x4_f32(false, a, false, b3, (short)0, acc3, false, false);
  }

#pragma unroll
  for (int r = 0; r < 8; ++r) {
    float* row = Y + (size_t)(mBase + r + hi * 8) * D;
    row[c0 +  0] = acc0[r];
    row[c0 + 16] = acc1[r];
    row[c0 + 32] = acc2[r];
    row[c0 + 48] = acc3[r];
  }
}

// ---------------------------------------------------------------------------
// Helpers
// ---------------------------------------------------------------------------
__device__ __forceinline__ float leaky02(float x) { return x > 0.0f ? x : 0.2f * x; }

// order-preserving float <-> uint encoding for atomicMax over signed floats
__device__ __forceinline__ unsigned fenc(float f) {
  unsigned u = __float_as_uint(f);
  return (u & 0x80000000u) ? ~u : (u | 0x80000000u);
}
__device__ __forceinline__ float fdec(unsigned u) {
  unsigned b = (u & 0x80000000u) ? (u & 0x7FFFFFFFu) : ~u;
  return __uint_as_float(b);
}

// ---------------------------------------------------------------------------
// Degree (GCN normalization).  dinv starts at 1.0 (self-loop), atomics add
// per incoming edge, then rsqrt in place.
// ---------------------------------------------------------------------------
__global__ void deg_init(float* __restrict__ dinv, int n) {
  int i = blockIdx.x * blockDim.x + threadIdx.x;
  if (i < n) dinv[i] = 1.0f;
}
__global__ void deg_edges(const int* __restrict__ ei, int e, float* __restrict__ dinv) {
  int i = blockIdx.x * blockDim.x + threadIdx.x;
  if (i < e) atomicAdd(&dinv[ei[e + i]], 1.0f);   // dst row of edge_index
}
__global__ void deg_rsqrt(float* __restrict__ dinv, int n) {
  int i = blockIdx.x * blockDim.x + threadIdx.x;
  if (i < n) dinv[i] = rsqrtf(dinv[i]);           // deg >= 1 always
}

// ---------------------------------------------------------------------------
// GCN aggregation
// ---------------------------------------------------------------------------
__global__ void gcn_self(const float* __restrict__ xw, const float* __restrict__ dinv,
                         float* __restrict__ agg) {
  int n = blockIdx.x, d = threadIdx.x;
  float di = dinv[n];
  agg[(size_t)n * D + d] = xw[(size_t)n * D + d] * di * di;
}
// one wave per edge; each lane moves 8 features (2 x float4)
__global__ void __launch_bounds__(256)
gcn_edges(const int* __restrict__ ei, int e, const float* __restrict__ xw,
          const float* __restrict__ dinv, float* __restrict__ agg) {
  int edge = blockIdx.x * 8 + (threadIdx.x >> 5);
  if (edge >= e) return;
  int s = ei[edge], t = ei[e + edge];
  float norm = dinv[s] * dinv[t];
  int lane = threadIdx.x & 31;
  const float4* xs = (const float4*)(xw + (size_t)s * D);
  float* ad = agg + (size_t)t * D;
#pragma unroll
  for (int i = 0; i < 2; ++i) {
    int q = lane + 32 * i;
    float4 v = xs[q];
    atomicAdd(&ad[q * 4 + 0], v.x * norm);
    atomicAdd(&ad[q * 4 + 1], v.y * norm);
    atomicAdd(&ad[q * 4 + 2], v.z * norm);
    atomicAdd(&ad[q * 4 + 3], v.w * norm);
  }
}
__global__ void bias_relu_inplace(float* __restrict__ x, const float* __restrict__ b) {
  int n = blockIdx.x, d = threadIdx.x;
  size_t idx = (size_t)n * D + d;
  x[idx] = fmaxf(x[idx] + b[d], 0.0f);
}

// ---------------------------------------------------------------------------
// GAT attention
// ---------------------------------------------------------------------------
// wave per node: aS[n] = <xw2[n], att_src>, aD[n] = <xw2[n], att_dst>
__global__ void __launch_bounds__(256)
gat_dots(const float* __restrict__ xw, const float* __restrict__ attS,
         const float* __restrict__ attD, float* __restrict__ aS,
         float* __restrict__ aD, int n) {
  int node = blockIdx.x * 8 + (threadIdx.x >> 5);
  if (node >= n) return;
  int lane = threadIdx.x & 31;
  const float* xr = xw + (size_t)node * D;
  float ss = 0.0f, dd = 0.0f;
#pragma unroll
  for (int i = lane; i < D; i += 32) {
    float v = xr[i];
    ss += v * attS[i];
    dd += v * attD[i];
  }
#pragma unroll
  for (int off = 16; off > 0; off >>= 1) {
    ss += __shfl_down(ss, off, 32);
    dd += __shfl_down(dd, off, 32);
  }
  if (lane == 0) { aS[node] = ss; aD[node] = dd; }
}
__global__ void m_init(const float* __restrict__ aS, const float* __restrict__ aD,
                       unsigned* __restrict__ m, int n) {
  int i = blockIdx.x * blockDim.x + threadIdx.x;
  if (i < n) m[i] = fenc(leaky02(aS[i] + aD[i]));   // self-loop logit
}
__global__ void m_edges(const int* __restrict__ ei, int e, const float* __restrict__ aS,
                        const float* __restrict__ aD, unsigned* __restrict__ m) {
  int i = blockIdx.x * blockDim.x + threadIdx.x;
  if (i >= e) return;
  int s = ei[i], t = ei[e + i];
  atomicMax(&m[t], fenc(leaky02(aS[s] + aD[t])));
}
__global__ void s_init(const float* __restrict__ aS, const float* __restrict__ aD,
                       const unsigned* __restrict__ m, float* __restrict__ sum, int n) {
  int i = blockIdx.x * blockDim.x + threadIdx.x;
  if (i < n) sum[i] = __expf(leaky02(aS[i] + aD[i]) - fdec(m[i]));  // self-loop term
}
__global__ void s_edges(const int* __restrict__ ei, int e, const float* __restrict__ aS,
                        const float* __restrict__ aD, const unsigned* __restrict__ m,
                        float* __restrict__ sum, float* __restrict__ exbuf) {
  int i = blockIdx.x * blockDim.x + threadIdx.x;
  if (i >= e) return;
  int s = ei[i], t = ei[e + i];
  float ex = __expf(leaky02(aS[s] + aD[t]) - fdec(m[t]));
  exbuf[i] = ex;
  atomicAdd(&sum[t], ex);
}
// out = alpha_self * xw2  (self-loop contribution seeds the accumulator)
__global__ void gat_self(const float* __restrict__ xw, const float* __restrict__ aS,
                         const float* __restrict__ aD, const unsigned* __restrict__ m,
                         const float* __restrict__ sum, float* __restrict__ out) {
  int n = blockIdx.x, d = threadIdx.x;
  float alpha = __expf(leaky02(aS[n] + aD[n]) - fdec(m[n])) / (sum[n] + 1e-16f);
  out[(size_t)n * D + d] = alpha * xw[(size_t)n * D + d];
}
__global__ void __launch_bounds__(256)
gat_edges(const int* __restrict__ ei, int e, const float* __restrict__ xw,
          const float* __restrict__ exbuf, const float* __restrict__ sum,
          float* __restrict__ out) {
  int edge = blockIdx.x * 8 + (threadIdx.x >> 5);
  if (edge >= e) return;
  int s = ei[edge], t = ei[e + edge];
  float alpha = exbuf[edge] / (sum[t] + 1e-16f);
  int lane = threadIdx.x & 31;
  const float4* xs = (const float4*)(xw + (size_t)s * D);
  float* od = out + (size_t)t * D;
#pragma unroll
  for (int i = 0; i < 2; ++i) {
    int q = lane + 32 * i;
    float4 v = xs[q];
    atomicAdd(&od[q * 4 + 0], v.x * alpha);
    atomicAdd(&od[q * 4 + 1], v.y * alpha);
    atomicAdd(&od[q * 4 + 2], v.z * alpha);
    atomicAdd(&od[q * 4 + 3], v.w * alpha);
  }
}

// ---------------------------------------------------------------------------
// Launch
// ---------------------------------------------------------------------------
extern "C" void kernel_launch(void* const* d_in, const int* in_sizes, int n_in,
                              void* d_out, int out_size, void* d_ws, size_t ws_size,
                              hipStream_t stream) {
  const float* x0   = (const float*)d_in[0];
  const int*   ei   = (const int*)d_in[1];    // [2,E] row-major: src row then dst row
  const float* W1   = (const float*)d_in[2];
  const float* b1   = (const float*)d_in[3];
  const float* W2   = (const float*)d_in[4];
  const float* attS = (const float*)d_in[5];
  const float* attD = (const float*)d_in[6];
  const float* b2   = (const float*)d_in[7];
  const int N = in_sizes[0] / D;              // 10000 (multiple of 16)
  const int E = in_sizes[1] / 2;              // 320000
  float* out = (float*)d_out;

  // workspace carve-up (all 4-byte elements)
  float*    bufA  = (float*)d_ws;             // N*D : xw1, later xw2
  float*    bufB  = bufA + (size_t)N * D;     // N*D : agg1 -> x1
  float*    dinv  = bufB + (size_t)N * D;     // N
  float*    aS    = dinv + N;                 // N
  float*    aD    = aS + N;                   // N
  unsigned* mEnc  = (unsigned*)(aD + N);      // N
  float*    sArr  = (float*)(mEnc + N);       // N
  float*    exbuf = sArr + N;                 // E

  const int TPB = 256;
  const int nBlk  = (N + TPB - 1) / TPB;
  const int eBlk  = (E + TPB - 1) / TPB;
  const int eWave = (E + 7) / 8;
  const int nWave = (N + 7) / 8;

  // ---- GCN ----
  gemm256_wmma_f32<<<N / 16, 128, 0, stream>>>(x0, W1, bufA);            // xw1
  deg_init <<<nBlk, TPB, 0, stream>>>(dinv, N);
  deg_edges<<<eBlk, TPB, 0, stream>>>(ei, E, dinv);
  deg_rsqrt<<<nBlk, TPB, 0, stream>>>(dinv, N);
  gcn_self <<<N, D, 0, stream>>>(bufA, dinv, bufB);
  gcn_edges<<<eWave, TPB, 0, stream>>>(ei, E, bufA, dinv, bufB);
  bias_relu_inplace<<<N, D, 0, stream>>>(bufB, b1);                      // x1

  // ---- GAT ----
  gemm256_wmma_f32<<<N / 16, 128, 0, stream>>>(bufB, W2, bufA);          // xw2
  gat_dots<<<nWave, TPB, 0, stream>>>(bufA, attS, attD, aS, aD, N);
  m_init  <<<nBlk, TPB, 0, stream>>>(aS, aD, mEnc, N);
  m_edges <<<eBlk, TPB, 0, stream>>>(ei, E, aS, aD, mEnc);
  s_init  <<<nBlk, TPB, 0, stream>>>(aS, aD, mEnc, sArr, N);
  s_edges <<<eBlk, TPB, 0, stream>>>(ei, E, aS, aD, mEnc, sArr, exbuf);
  gat_self<<<N, D, 0, stream>>>(bufA, aS, aD, mEnc, sArr, out);
  gat_edges<<<eWave, TPB, 0, stream>>>(ei, E, bufA, exbuf, sArr, out);
  bias_relu_inplace<<<N, D, 0, stream>>>(out, b2);
}